// WaveletConv2d_64390149702241
// MI455X (gfx1250) — compile-verified
//
#include <hip/hip_runtime.h>

typedef __attribute__((ext_vector_type(2))) float v2f;
typedef __attribute__((ext_vector_type(8))) float v8f;

// db6 decomposition low-pass (float64 source truncated to f32)
__constant__ float F_DEC_LO[12] = {
  -0.00107730108499558f,  0.004777257511010651f, 0.0005538422009938016f,
  -0.031582039318031156f, 0.02752286553001629f,  0.09750160558707936f,
  -0.12976686756709563f, -0.22626469396516913f,  0.3152503517092432f,
   0.7511339080215775f,   0.4946238903983854f,   0.11154074335008017f };

// A_LO = rec_lo = reverse(dec_lo); A_HI[t] = dec_lo[t] * (t odd ? -1 : +1)
// S_LO = dec_lo;                   S_HI[t] = dec_lo[11-t] * (t odd ? +1 : -1)
__device__ __forceinline__ float f_a_lo(int t){ return F_DEC_LO[11 - t]; }
__device__ __forceinline__ float f_a_hi(int t){ float v = F_DEC_LO[t];      return (t & 1) ? -v : v; }
__device__ __forceinline__ float f_s_lo(int t){ return F_DEC_LO[t]; }
__device__ __forceinline__ float f_s_hi(int t){ float v = F_DEC_LO[11 - t]; return (t & 1) ?  v : -v; }

// ---------------- Analysis filter bank along last axis (W) ----------------
// in: [rows, Win] -> lo,hi: [rows, Wout], symmetric pad, padL = 10, stride 2
__global__ void afb_w_kernel(const float* __restrict__ in, float* __restrict__ lo,
                             float* __restrict__ hi, long rows, int Win, int Wout)
{
  long g = blockIdx.x * (long)blockDim.x + threadIdx.x;
  long total = rows * (long)Wout;
  if (g >= total) return;
  int  m = (int)(g % Wout);
  long r = g / Wout;
  const float* row = in + r * (long)Win;
  float accLo = 0.f, accHi = 0.f;
#pragma unroll
  for (int t = 0; t < 12; ++t) {
    int idx = 2 * m + t - 10;
    idx = (idx < 0) ? (-idx - 1) : idx;
    idx = (idx >= Win) ? (2 * Win - 1 - idx) : idx;
    float v = row[idx];
    accLo += v * f_a_lo(t);
    accHi += v * f_a_hi(t);
  }
  lo[g] = accLo;
  hi[g] = accHi;
}

// ---------------- Analysis filter bank along H ----------------
// in: [BC, Hin, W] -> lo,hi: [BC, Hout, W]
__global__ void afb_h_kernel(const float* __restrict__ in, float* __restrict__ lo,
                             float* __restrict__ hi, int BC, int Hin, int Hout, int W)
{
  long g = blockIdx.x * (long)blockDim.x + threadIdx.x;
  long total = (long)BC * Hout * W;
  if (g >= total) return;
  int  w  = (int)(g % W);
  long t1 = g / W;
  int  m  = (int)(t1 % Hout);
  int  bc = (int)(t1 / Hout);
  const float* base = in + (long)bc * Hin * W + w;
  float accLo = 0.f, accHi = 0.f;
#pragma unroll
  for (int t = 0; t < 12; ++t) {
    int idx = 2 * m + t - 10;
    idx = (idx < 0) ? (-idx - 1) : idx;
    idx = (idx >= Hin) ? (2 * Hin - 1 - idx) : idx;
    float v = base[(long)idx * W];
    accLo += v * f_a_lo(t);
    accHi += v * f_a_hi(t);
  }
  lo[g] = accLo;
  hi[g] = accHi;
}

// ---------------- Synthesis filter bank along H ----------------
// lo (strided, allows crop): element (bc,j,w) at bc*loBC + j*loRow + w
// hi: [BC, n, W] contiguous. out: [BC, Hout, W], Hout = 2n-10.
__global__ void sfb_h_kernel(const float* __restrict__ lo, const float* __restrict__ hi,
                             float* __restrict__ out, int BC, int n, int W, int Hout,
                             int loRow, long loBC)
{
  long g = blockIdx.x * (long)blockDim.x + threadIdx.x;
  long total = (long)BC * Hout * W;
  if (g >= total) return;
  int  w  = (int)(g % W);
  long t1 = g / W;
  int  m  = (int)(t1 % Hout);
  int  bc = (int)(t1 / Hout);
  const float* lop = lo + (long)bc * loBC + w;
  const float* hip = hi + (long)bc * n * W + w;
  int j0   = m >> 1;
  int tap0 = (m & 1) ? 0 : 1;   // even m -> odd taps; odd m -> even taps
  float acc = 0.f;
#pragma unroll
  for (int s = 0; s < 6; ++s) {
    int t = 2 * s + tap0;
    int j = j0 + s;
    acc += lop[(long)j * loRow] * f_s_lo(t) + hip[(long)j * W] * f_s_hi(t);
  }
  out[g] = acc;
}

// ---------------- Synthesis filter bank along W ----------------
// lo,hi: [rows, n] -> out: [rows, Wout], Wout = 2n-10
__global__ void sfb_w_kernel(const float* __restrict__ lo, const float* __restrict__ hi,
                             float* __restrict__ out, long rows, int n, int Wout)
{
  long g = blockIdx.x * (long)blockDim.x + threadIdx.x;
  long total = rows * (long)Wout;
  if (g >= total) return;
  int  m = (int)(g % Wout);
  long r = g / Wout;
  const float* lop = lo + r * (long)n;
  const float* hip = hi + r * (long)n;
  int j0   = m >> 1;
  int tap0 = (m & 1) ? 0 : 1;
  float acc = 0.f;
#pragma unroll
  for (int s = 0; s < 6; ++s) {
    int t = 2 * s + tap0;
    int j = j0 + s;
    acc += lop[j] * f_s_lo(t) + hip[j] * f_s_hi(t);
  }
  out[g] = acc;
}

// ---------------- Per-pixel channel mix via f32 WMMA, subband-batched ----------------
// d0..d3: up to 4 subband tensors [B=8, C=64, S, S], mixed IN PLACE at pixel p:
//   out[sub][b,o,p] = sum_i in[sub][b,i,p] * Wint[i,o,p]
// All subbands at one level share the same per-pixel bilinear weight tile, so one
// block (256 thr = 8 wave32) handles one pixel across all nsub subbands:
//   A matrix M = 32 rows (sub*8 + batch; zero-padded past nsub*8), K = 64, N = 64.
// Wave (mt,nt) computes M-tile mt (16 rows) x N-tile nt (16 cols) with 16 chained
// V_WMMA_F32_16X16X4_F32.
__global__ void mix4_kernel(float* __restrict__ d0, float* __restrict__ d1,
                            float* __restrict__ d2, float* __restrict__ d3,
                            int nsub, const float* __restrict__ weights,
                            int S, long P)
{
  __shared__ float sW[64 * 64];   // interpolated weight tile [K=cin][N=cout] (16 KB)
  __shared__ float sA[32 * 64];   // activations [M=sub*8+b (32)][K=cin]       (8 KB)

  int p = blockIdx.x;
  int y = p / S, x = p % S;
  float scale = 64.0f / (float)S;
  float fy = fminf(fmaxf(((float)y + 0.5f) * scale - 0.5f, 0.0f), 63.0f);
  float fx = fminf(fmaxf(((float)x + 0.5f) * scale - 0.5f, 0.0f), 63.0f);
  int y0 = (int)floorf(fy); int y1 = (y0 + 1 < 63) ? (y0 + 1) : 63;
  int x0 = (int)floorf(fx); int x1 = (x0 + 1 < 63) ? (x0 + 1) : 63;
  float ty = fy - (float)y0, tx = fx - (float)x0;

  int tid = threadIdx.x;
  // Phase 1: bilinear-blend 4 corner 64x64 matrices into LDS (corners stay L2-resident)
  for (int idx = tid; idx < 4096; idx += 256) {
    const float* wp = weights + ((long)idx << 12);  // idx = i*64+o, stride 64*64
    float w00 = wp[y0 * 64 + x0], w01 = wp[y0 * 64 + x1];
    float w10 = wp[y1 * 64 + x0], w11 = wp[y1 * 64 + x1];
    sW[idx] = (w00 * (1.f - tx) + w01 * tx) * (1.f - ty)
            + (w10 * (1.f - tx) + w11 * tx) * ty;
  }
  // Phase 2: load this pixel's activations for all subbands (pad rows past nsub*8)
  for (int idx = tid; idx < 2048; idx += 256) {
    int m = idx >> 6, i = idx & 63;
    int sub = m >> 3, b = m & 7;
    float v = 0.f;
    if (sub < nsub) {
      const float* d = (sub == 0) ? d0 : (sub == 1) ? d1 : (sub == 2) ? d2 : d3;
      v = d[(long)(b * 64 + i) * P + p];
    }
    sA[idx] = v;
  }
  __syncthreads();

  int wave = tid >> 5, lane = tid & 31;
  int mt = wave >> 2, nt = wave & 3;      // 2 M-tiles x 4 N-tiles
  int n0   = nt * 16;
  int half = lane >> 4;     // 0: lanes 0-15 (K+0/1), 1: lanes 16-31 (K+2/3)
  int l16  = lane & 15;
  int arow = (mt * 16 + l16) * 64;

  v8f acc = {0.f, 0.f, 0.f, 0.f, 0.f, 0.f, 0.f, 0.f};
#pragma unroll
  for (int k0 = 0; k0 < 64; k0 += 4) {
    int ka = k0 + half * 2;
    v2f a = { sA[arow + ka],              sA[arow + ka + 1] };            // A[M][K]
    v2f b = { sW[ka * 64 + n0 + l16],     sW[(ka + 1) * 64 + n0 + l16] }; // B[K][N]
    acc = __builtin_amdgcn_wmma_f32_16x16x4_f32(false, a, false, b,
                                                (short)0, acc, false, false);
  }

  // D layout: VGPR r -> M = mt*16 + r (lanes 0-15) / + r+8 (lanes 16-31), N = n0+l16.
#pragma unroll
  for (int r = 0; r < 8; ++r) {
    int M = mt * 16 + half * 8 + r;
    int sub = M >> 3, b = M & 7;
    if (sub < nsub) {
      float* d = (sub == 0) ? d0 : (sub == 1) ? d1 : (sub == 2) ? d2 : d3;
      d[(long)(b * 64 + n0 + l16) * P + p] = acc[r];
    }
  }
}

static inline unsigned nblk(long total) { return (unsigned)((total + 255) / 256); }

extern "C" void kernel_launch(void* const* d_in, const int* in_sizes, int n_in,
                              void* d_out, int out_size, void* d_ws, size_t ws_size,
                              hipStream_t stream) {
  (void)in_sizes; (void)n_in; (void)out_size; (void)ws_size;
  const float* x   = (const float*)d_in[0];   // [8,64,256,256]
  const float* wts = (const float*)d_in[1];   // [64,64,64,64]
  float* out = (float*)d_out;                 // [8,64,256,256]
  float* ws  = (float*)d_ws;

  const int BC = 512;              // B*C
  const int H0 = 256, S1 = 133, S2 = 72, S3 = 41;

  // Workspace carve-up (floats). Row-transform scratch doubles as IDWT temps.
  const size_t SZ_ROW = (size_t)BC * H0 * S1;       // 17,432,576 (max temp)
  const size_t SZ_Y1  = (size_t)BC * 134 * 134;     // 9,193,472  (>= BC*S1*S1)
  const size_t SZ1    = (size_t)BC * S1 * S1;
  const size_t SZ2    = (size_t)BC * S2 * S2;
  const size_t SZ3    = (size_t)BC * S3 * S3;

  float* rowA = ws;
  float* rowB = rowA + SZ_ROW;
  float* ll1  = rowB + SZ_ROW;     // later reused for y1 [BC,134,134]
  float* lh1  = ll1 + SZ_Y1;
  float* hl1  = lh1 + SZ1;
  float* hh1  = hl1 + SZ1;
  float* ll2  = hh1 + SZ1;         // later reused for y2 [BC,72,72]
  float* lh2  = ll2 + SZ2;
  float* hl2  = lh2 + SZ2;
  float* hh2  = hl2 + SZ2;
  float* ll3  = hh2 + SZ2;
  float* lh3  = ll3 + SZ3;
  float* hl3  = lh3 + SZ3;
  float* hh3  = hl3 + SZ3;

  dim3 thr(256);

  // ---- Forward DWT level 1: 256 -> 133 ----
  afb_w_kernel<<<nblk((long)BC * H0 * S1), thr, 0, stream>>>(x, rowA, rowB, (long)BC * H0, H0, S1);
  afb_h_kernel<<<nblk((long)BC * S1 * S1), thr, 0, stream>>>(rowA, ll1, lh1, BC, H0, S1, S1);
  afb_h_kernel<<<nblk((long)BC * S1 * S1), thr, 0, stream>>>(rowB, hl1, hh1, BC, H0, S1, S1);
  // ---- Level 2: 133 -> 72 ----
  afb_w_kernel<<<nblk((long)BC * S1 * S2), thr, 0, stream>>>(ll1, rowA, rowB, (long)BC * S1, S1, S2);
  afb_h_kernel<<<nblk((long)BC * S2 * S2), thr, 0, stream>>>(rowA, ll2, lh2, BC, S1, S2, S2);
  afb_h_kernel<<<nblk((long)BC * S2 * S2), thr, 0, stream>>>(rowB, hl2, hh2, BC, S1, S2, S2);
  // ---- Level 3: 72 -> 41 ----
  afb_w_kernel<<<nblk((long)BC * S2 * S3), thr, 0, stream>>>(ll2, rowA, rowB, (long)BC * S2, S2, S3);
  afb_h_kernel<<<nblk((long)BC * S3 * S3), thr, 0, stream>>>(rowA, ll3, lh3, BC, S2, S3, S3);
  afb_h_kernel<<<nblk((long)BC * S3 * S3), thr, 0, stream>>>(rowB, hl3, hh3, BC, S2, S3, S3);

  // ---- Per-pixel channel mixing (in place, f32 WMMA, subbands batched per level) ----
  long P3 = (long)S3 * S3, P2 = (long)S2 * S2, P1 = (long)S1 * S1;
  mix4_kernel<<<(unsigned)P3, 256, 0, stream>>>(ll3, lh3, hl3, hh3, 4, wts, S3, P3);
  mix4_kernel<<<(unsigned)P2, 256, 0, stream>>>(lh2, hl2, hh2, hh2, 3, wts, S2, P2);
  mix4_kernel<<<(unsigned)P1, 256, 0, stream>>>(lh1, hl1, hh1, hh1, 3, wts, S1, P1);

  // ---- Inverse DWT level 3: 41 -> 72 (y2 -> ll2 buffer) ----
  sfb_h_kernel<<<nblk((long)BC * S2 * S3), thr, 0, stream>>>(ll3, lh3, rowA, BC, S3, S3, S2, S3, (long)S3 * S3);
  sfb_h_kernel<<<nblk((long)BC * S2 * S3), thr, 0, stream>>>(hl3, hh3, rowB, BC, S3, S3, S2, S3, (long)S3 * S3);
  sfb_w_kernel<<<nblk((long)BC * S2 * S2), thr, 0, stream>>>(rowA, rowB, ll2, (long)BC * S2, S3, S2);
  // ---- Level 2: 72 -> 134 (y1 -> ll1 buffer) ----
  sfb_h_kernel<<<nblk((long)BC * 134 * S2), thr, 0, stream>>>(ll2, lh2, rowA, BC, S2, S2, 134, S2, (long)S2 * S2);
  sfb_h_kernel<<<nblk((long)BC * 134 * S2), thr, 0, stream>>>(hl2, hh2, rowB, BC, S2, S2, 134, S2, (long)S2 * S2);
  sfb_w_kernel<<<nblk((long)BC * 134 * 134), thr, 0, stream>>>(rowA, rowB, ll1, (long)BC * 134, S2, 134);
  // ---- Level 1: crop y1 134->133, 133 -> 256 (final -> d_out) ----
  sfb_h_kernel<<<nblk((long)BC * H0 * S1), thr, 0, stream>>>(ll1, lh1, rowA, BC, S1, S1, H0, 134, (long)134 * 134);
  sfb_h_kernel<<<nblk((long)BC * H0 * S1), thr, 0, stream>>>(hl1, hh1, rowB, BC, S1, S1, H0, S1, (long)S1 * S1);
  sfb_w_kernel<<<nblk((long)BC * H0 * H0), thr, 0, stream>>>(rowA, rowB, out, (long)BC * H0, S1, H0);
}